// NeRFRenderer_32409823215678
// MI455X (gfx1250) — compile-verified
//
#include <hip/hip_runtime.h>
#include <math.h>

// NeRF hierarchical sampling, N=262144 rays x 128 samples. Bandwidth-bound
// (~1.2 GB @ 23.3 TB/s => ~50us floor). One wave32 per ray, 4 samples/lane.
// CDNA5 paths: global_load_async_to_lds_b128 double-buffered sigma pipeline
// (ASYNCcnt), NT-hinted b128 stores (1GB write stream vs 192MB L2), wave32
// shuffle scans, LDS CDF binary search, rcp/exp native VALU.

typedef float v4f __attribute__((ext_vector_type(4)));

#define NS 128
#define NU 128

__device__ __forceinline__ float clampf(float x, float lo, float hi) {
    return __builtin_amdgcn_fmed3f(x, lo, hi);
}

__global__ __launch_bounds__(256) void nerf_render_kernel(
    const float* __restrict__ rays_o,
    const float* __restrict__ rays_d,
    const float* __restrict__ sigmas,
    float* __restrict__ out_xyzs,   // [N, NS, 3]
    float* __restrict__ out_w,      // [N, NS]
    float* __restrict__ out_nz,     // [N, NU]
    float* __restrict__ out_nxyz,   // [N, NU, 3]
    int N, int nwaves)
{
    __shared__ float s_cum[8][NS];          // per-wave CDF        (4 KB)
    __shared__ float s_sig[8][2][NS];       // async double buffer (8 KB)
    const int lane = threadIdx.x & 31;
    const int wid  = threadIdx.x >> 5;
    float* cum = s_cum[wid];
    const int wave0 = (int)blockIdx.x * 8 + wid;

    const float inv127 = 1.0f / 127.0f;
    const float inv128 = 1.0f / 128.0f;

    // LDS byte offsets for the async engine (generic ptr low 32 bits = LDS addr)
    const unsigned myoff = (unsigned)lane * 16u;                     // global row offset
    const unsigned lbase = (unsigned)(uintptr_t)(&s_sig[wid][0][0]) + myoff;

    // ---- pipeline prologue: async-load first ray's sigma row into buf 0 ----
    if (wave0 < N) {
        const float* p0 = sigmas + (size_t)wave0 * NS;
        asm volatile("global_load_async_to_lds_b128 %0, %1, %2"
                     :: "v"(lbase), "v"(myoff), "s"(p0) : "memory");
    }
    int buf = 0;

    for (int ray = wave0; ray < N; ray += nwaves) {
        // ---- issue next ray's async sigma load, then wait for current ----
        int nray = ray + nwaves;
        if (nray < N) {
            const float* pn = sigmas + (size_t)nray * NS;
            unsigned dst = lbase + (unsigned)((buf ^ 1) * (NS * 4));
            asm volatile("global_load_async_to_lds_b128 %0, %1, %2"
                         :: "v"(dst), "v"(myoff), "s"(pn) : "memory");
            asm volatile("s_wait_asynccnt 0x1" ::: "memory");
        } else {
            asm volatile("s_wait_asynccnt 0x0" ::: "memory");
        }

        // ---- ray data (wave-uniform -> scalarizable) ----
        float ox = rays_o[3 * ray + 0], oy = rays_o[3 * ray + 1], oz = rays_o[3 * ray + 2];
        float dx = rays_d[3 * ray + 0], dy = rays_d[3 * ray + 1], dz = rays_d[3 * ray + 2];

        // ---- near/far vs AABB lo=(-0.5,-0.1,-0.55) hi=(0.8,0.5,0.55) ----
        // both slab planes share a denominator: 3 x v_rcp_f32 + 6 muls
        float rx = __builtin_amdgcn_rcpf(dx + 1e-15f);
        float ry = __builtin_amdgcn_rcpf(dy + 1e-15f);
        float rz = __builtin_amdgcn_rcpf(dz + 1e-15f);
        float t0x = (-0.5f  - ox) * rx, t1x = (0.8f  - ox) * rx;
        float t0y = (-0.1f  - oy) * ry, t1y = (0.5f  - oy) * ry;
        float t0z = (-0.55f - oz) * rz, t1z = (0.55f - oz) * rz;
        float nxm = t0x < t1x ? t0x : t1x;
        float nym = t0y < t1y ? t0y : t1y;
        float nzm = t0z < t1z ? t0z : t1z;
        float fxm = t0x > t1x ? t0x : t1x;
        float fym = t0y > t1y ? t0y : t1y;
        float fzm = t0z > t1z ? t0z : t1z;
        float nearv = fmaxf(fmaxf(nxm, nym), nzm);
        float farv  = fminf(fminf(fxm, fym), fzm);
        if (farv < nearv) { nearv = 1e9f; farv = 1e9f; }
        nearv = fmaxf(nearv, 0.05f);
        float span  = farv - nearv;
        float dmain = span * inv127;    // delta for s < 127
        float dlast = span * inv128;    // sample_dist at s == 127

        // ---- sigmas from the async-filled LDS buffer ----
        const v4f* sp = (const v4f*)(&s_sig[wid][buf][0]);
        v4f sg = sp[lane];

        // ---- alphas + transmittance terms, samples s = 4*lane + j ----
        float a[4], m[4];
        #pragma unroll
        for (int j = 0; j < 4; ++j) {
            int s = 4 * lane + j;
            float dl = (s == NS - 1) ? dlast : dmain;
            a[j] = 1.0f - __expf(-dl * sg[j]);        // v_exp_f32
            m[j] = 1.0f - a[j] + 1e-15f;
        }

        // ---- wave-wide exclusive prefix product (transmittance) ----
        float inc = m[0] * m[1] * m[2] * m[3];
        #pragma unroll
        for (int off = 1; off < 32; off <<= 1) {
            float other = __shfl_up(inc, off, 32);
            if (lane >= off) inc *= other;
        }
        float T = __shfl_up(inc, 1, 32);
        if (lane == 0) T = 1.0f;

        float w[4];
        #pragma unroll
        for (int j = 0; j < 4; ++j) { w[j] = a[j] * T; T *= m[j]; }

        // ---- coarse positions + NT streamed stores ----
        float px[4], py[4], pz[4];
        #pragma unroll
        for (int j = 0; j < 4; ++j) {
            int s = 4 * lane + j;
            float z = nearv + span * ((float)s * inv127);
            px[j] = clampf(ox + dx * z, -0.5f,  0.8f);
            py[j] = clampf(oy + dy * z, -0.1f,  0.5f);
            pz[j] = clampf(oz + dz * z, -0.55f, 0.55f);
        }
        {
            v4f q0 = {px[0], py[0], pz[0], px[1]};
            v4f q1 = {py[1], pz[1], px[2], py[2]};
            v4f q2 = {pz[2], px[3], py[3], pz[3]};
            v4f* dst = (v4f*)(out_xyzs + (size_t)ray * NS * 3) + lane * 3;
            __builtin_nontemporal_store(q0, dst + 0);
            __builtin_nontemporal_store(q1, dst + 1);
            __builtin_nontemporal_store(q2, dst + 2);
            v4f wv = {w[0], w[1], w[2], w[3]};
            __builtin_nontemporal_store(wv, (v4f*)(out_w + (size_t)ray * NS) + lane);
        }

        // ---- unnormalized CDF: q[s] = (1<=s<=126) ? w[s]+1e-5 : 0 ----
        float run = 0.0f, cums[4];
        #pragma unroll
        for (int j = 0; j < 4; ++j) {
            int s = 4 * lane + j;
            float qv = (s >= 1 && s <= 126) ? (w[j] + 1e-5f) : 0.0f;
            run += qv;
            cums[j] = run;
        }
        float incS = run;
        #pragma unroll
        for (int off = 1; off < 32; off <<= 1) {
            float other = __shfl_up(incS, off, 32);
            if (lane >= off) incS += other;
        }
        float excl = __shfl_up(incS, 1, 32);
        if (lane == 0) excl = 0.0f;
        float total = __shfl(incS, 31, 32);

        #pragma unroll
        for (int j = 0; j < 4; ++j) {
            int s = 4 * lane + j;
            if (s <= 126) cum[s] = excl + cums[j];    // cdf indices 0..126, cum[0]==0
        }
        __builtin_amdgcn_wave_barrier();   // same-wave DS ops are in-order; fence compiler

        // ---- sample_pdf (det): u_k=(k+0.5)/128, search unnormalized cum ----
        float inv_total = __builtin_amdgcn_rcpf(total);
        float nzv[4], qx[4], qy[4], qz[4];
        #pragma unroll
        for (int j = 0; j < 4; ++j) {
            int k = 4 * lane + j;
            float u = ((float)k + 0.5f) * inv128;
            float target = u * total;
            int lo = 0, hi = 127;                 // first i in [0,127) with cdf[i] > u
            #pragma unroll
            for (int it = 0; it < 7; ++it) {
                int mid = (lo + hi) >> 1;
                bool gt = cum[mid] > target;
                hi = gt ? mid : hi;
                lo = gt ? lo : (mid + 1);
            }
            int below = (lo - 1 < 0) ? 0 : lo - 1;
            int above = (lo > 126) ? 126 : lo;
            float cb = cum[below] * inv_total;
            float ca = cum[above] * inv_total;
            float den = ca - cb;
            den = (den < 1e-5f) ? 1.0f : den;
            float tt = (u - cb) * __builtin_amdgcn_rcpf(den);
            // bins affine in index: z_mid[i] = near + span*(i+0.5)/127
            float bb = nearv + span * (((float)below + 0.5f) * inv127);
            float ba = nearv + span * (((float)above + 0.5f) * inv127);
            float znew = bb + tt * (ba - bb);
            nzv[j] = znew;
            qx[j] = clampf(ox + dx * znew, -0.5f,  0.8f);
            qy[j] = clampf(oy + dy * znew, -0.1f,  0.5f);
            qz[j] = clampf(oz + dz * znew, -0.55f, 0.55f);
        }
        {
            v4f nzq = {nzv[0], nzv[1], nzv[2], nzv[3]};
            __builtin_nontemporal_store(nzq, (v4f*)(out_nz + (size_t)ray * NU) + lane);
            v4f q0 = {qx[0], qy[0], qz[0], qx[1]};
            v4f q1 = {qy[1], qz[1], qx[2], qy[2]};
            v4f q2 = {qz[2], qx[3], qy[3], qz[3]};
            v4f* dst = (v4f*)(out_nxyz + (size_t)ray * NU * 3) + lane * 3;
            __builtin_nontemporal_store(q0, dst + 0);
            __builtin_nontemporal_store(q1, dst + 1);
            __builtin_nontemporal_store(q2, dst + 2);
        }
        __builtin_amdgcn_wave_barrier();   // next iter's LDS writes after this iter's reads
        buf ^= 1;
    }
}

extern "C" void kernel_launch(void* const* d_in, const int* in_sizes, int n_in,
                              void* d_out, int out_size, void* d_ws, size_t ws_size,
                              hipStream_t stream) {
    const float* rays_o = (const float*)d_in[0];
    const float* rays_d = (const float*)d_in[1];
    const float* sigmas = (const float*)d_in[2];
    // d_in[3]/d_in[4] are num_steps/upsample_steps, fixed at 128 by the reference.
    int N = in_sizes[0] / 3;

    float* out      = (float*)d_out;
    float* out_xyzs = out;                                  // N*NS*3
    float* out_w    = out_xyzs + (size_t)N * NS * 3;        // N*NS
    float* out_nz   = out_w    + (size_t)N * NS;            // N*NU
    float* out_nxyz = out_nz   + (size_t)N * NU;            // N*NU*3

    int blocks = 2048;                    // 16384 waves; 16 rays/wave at N=262144
    int maxBlocks = (N + 7) / 8;
    if (blocks > maxBlocks) blocks = maxBlocks;
    int nwaves = blocks * 8;

    nerf_render_kernel<<<blocks, 256, 0, stream>>>(
        rays_o, rays_d, sigmas, out_xyzs, out_w, out_nz, out_nxyz, N, nwaves);
}